// ErrorInterpolate_19645180412072
// MI455X (gfx1250) — compile-verified
//
#include <hip/hip_runtime.h>

// CDNA5 (gfx1250) wave32 WMMA types
typedef __attribute__((ext_vector_type(2))) float v2f;
typedef __attribute__((ext_vector_type(8))) float v8f;

#define KNN   3
#define EPSV  1e-16f
#define FINF  3.4028235e38f
#define NW    4          // waves per block
#define NT    128        // threads per block
#define MAXNL 4096       // coarse point capacity staged in LDS (48 KB)

// Branchless sorted top-3 insert of (d, j) into (t0<=t1<=t2).
#define TOP3_INSERT(d, j, t0d, t0i, t1d, t1i, t2d, t2i)                  \
    {                                                                    \
        const bool lt0 = (d) < (t0d);                                    \
        const bool lt1 = (d) < (t1d);                                    \
        const bool lt2 = (d) < (t2d);                                    \
        const float n2d = lt1 ? (t1d) : (lt2 ? (d) : (t2d));             \
        const int   n2i = lt1 ? (t1i) : (lt2 ? (j) : (t2i));             \
        const float n1d = lt0 ? (t0d) : (lt1 ? (d) : (t1d));             \
        const int   n1i = lt0 ? (t0i) : (lt1 ? (j) : (t1i));             \
        const float n0d = lt0 ? (d) : (t0d);                             \
        const int   n0i = lt0 ? (j) : (t0i);                             \
        (t0d) = n0d; (t1d) = n1d; (t2d) = n2d;                           \
        (t0i) = n0i; (t1i) = n1i; (t2i) = n2i;                           \
    }

// Block = 4 waves, 16 fine rows. pos_l staged to LDS via async copies.
// Each wave scans 1/4 of the coarse tiles with V_WMMA_F32_16X16X4_F32:
//   A = -2 * pos_h tile (16x4, K=3 padded), B = pos_l tile (4x16),
//   D = -2 h.l  ->  d2 = D + |h|^2 + |l|^2
__global__ __launch_bounds__(NT) void knn_interp_wmma(
    const float* __restrict__ x,      // [n_l, F]
    const float* __restrict__ pos_l,  // [n_l, 3]
    const float* __restrict__ pos_h,  // [n_h, 3]
    float* __restrict__ out,          // [n_h, F]
    int n_l, int n_h, int F)
{
    __shared__ float s_pos[MAXNL * 3];        // staged coarse positions (48 KB)
    __shared__ float s_md[NW][16][KNN];       // per-wave top-3 distances
    __shared__ int   s_mi[NW][16][KNN];       // per-wave top-3 indices
    __shared__ int   s_i[16][KNN];            // final indices
    __shared__ float s_w[16][4];              // w0,w1,w2, 1/sum(w)

    const int tid  = threadIdx.x;
    const int wave = tid >> 5;
    const int lane = tid & 31;
    const int lm   = lane & 15;
    const int half = lane >> 4;               // 0: K=0/1 + rows 0..7, 1: K=2/3 + rows 8..15
    const int rowBase = blockIdx.x * 16;

    // ---- Stage pos_l into LDS with CDNA5 async global->LDS copies ----
    {
        const int bytes = n_l * 12;           // 4096*12 = 49152, 16B-divisible
        const unsigned lbase = (unsigned)(uintptr_t)(&s_pos[0]); // LDS byte offset
        const uint64_t gbase = (uint64_t)(uintptr_t)pos_l;
        for (int ofs = tid * 16; ofs < bytes; ofs += NT * 16) {
            const unsigned laddr = lbase + (unsigned)ofs;
            const uint64_t gaddr = gbase + (uint64_t)ofs;
            asm volatile("global_load_async_to_lds_b128 %0, %1, off"
                         :: "v"(laddr), "v"(gaddr) : "memory");
        }
        asm volatile("s_wait_asynccnt 0" ::: "memory");
    }
    __syncthreads();

    // ---- A matrix (per-block constant): rows = fine points, scaled by -2 ----
    const int gm = rowBase + lm;
    const float hx = pos_h[3 * gm + 0];
    const float hy = pos_h[3 * gm + 1];
    const float hz = pos_h[3 * gm + 2];
    const float hn = hx * hx + hy * hy + hz * hz;
    v2f A;
    A.x = half ? (-2.0f * hz) : (-2.0f * hx); // VGPR0: K=0 (lanes 0-15) / K=2 (16-31)
    A.y = half ? 0.0f         : (-2.0f * hy); // VGPR1: K=1 / K=3 (pad)

    // |h|^2 for the 8 accumulator rows (M = r + 8*half); hn of row m is in lanes m, m+16.
    float hn_r[8];
#pragma unroll
    for (int r = 0; r < 8; ++r)
        hn_r[r] = __shfl(hn, r + (half << 3), 32);

    float td[8][KNN];
    int   ti[8][KNN];
#pragma unroll
    for (int r = 0; r < 8; ++r)
#pragma unroll
        for (int k = 0; k < KNN; ++k) { td[r][k] = FINF; ti[r][k] = 0; }

    // ---- Scan: this wave covers tiles t = wave, wave+NW, ... ----
    const int ntile = n_l >> 4;
    for (int t = wave; t < ntile; t += NW) {
        const int j = (t << 4) + lm;          // this lane's coarse column
        const float lx = s_pos[3 * j + 0];    // LDS reads (ds_load)
        const float ly = s_pos[3 * j + 1];
        const float lz = s_pos[3 * j + 2];
        const float ln = lx * lx + ly * ly + lz * lz;
        v2f B;
        B.x = half ? lz  : lx;                // VGPR0: K=0 / K=2
        B.y = half ? 0.f : ly;                // VGPR1: K=1 / K=3 (pad)

        v8f C = {};
        C = __builtin_amdgcn_wmma_f32_16x16x4_f32(
                /*neg_a=*/false, A, /*neg_b=*/false, B,
                /*c_mod=*/(short)0, C, /*reuse_a=*/false, /*reuse_b=*/false);

#pragma unroll
        for (int r = 0; r < 8; ++r) {
            const float d = C[r] + hn_r[r] + ln;
            TOP3_INSERT(d, j, td[r][0], ti[r][0], td[r][1], ti[r][1], td[r][2], ti[r][2]);
        }
    }

    // ---- XOR-butterfly merge within each 16-lane half of this wave ----
#pragma unroll
    for (int m = 1; m < 16; m <<= 1) {
#pragma unroll
        for (int r = 0; r < 8; ++r) {
            // snapshot partner values BEFORE any insertion (symmetric exchange)
            float bd[KNN]; int bi[KNN];
#pragma unroll
            for (int k = 0; k < KNN; ++k) {
                bd[k] = __shfl_xor(td[r][k], m, 32);
                bi[k] = __shfl_xor(ti[r][k], m, 32);
            }
#pragma unroll
            for (int k = 0; k < KNN; ++k) {
                TOP3_INSERT(bd[k], bi[k],
                            td[r][0], ti[r][0], td[r][1], ti[r][1], td[r][2], ti[r][2]);
            }
        }
    }

    // ---- Publish this wave's top-3 per row ----
#pragma unroll
    for (int r = 0; r < 8; ++r) {
        if (lm == r) {                         // one lane per half writes its row
            const int row = r + (half << 3);
#pragma unroll
            for (int k = 0; k < KNN; ++k) {
                s_md[wave][row][k] = td[r][k];
                s_mi[wave][row][k] = ti[r][k];
            }
        }
    }
    __syncthreads();

    // ---- Cross-wave merge + exact d2 recompute + weights (16 threads) ----
    if (tid < 16) {
        const int row = tid;
        float t0d = s_md[0][row][0], t1d = s_md[0][row][1], t2d = s_md[0][row][2];
        int   t0i = s_mi[0][row][0], t1i = s_mi[0][row][1], t2i = s_mi[0][row][2];
#pragma unroll
        for (int w = 1; w < NW; ++w)
#pragma unroll
            for (int k = 0; k < KNN; ++k)
                TOP3_INSERT(s_md[w][row][k], s_mi[w][row][k],
                            t0d, t0i, t1d, t1i, t2d, t2i);

        const int g = rowBase + row;
        const float px = pos_h[3 * g + 0];
        const float py = pos_h[3 * g + 1];
        const float pz = pos_h[3 * g + 2];
        const int idx3[KNN] = { t0i, t1i, t2i };
        float wsum = 0.0f;
#pragma unroll
        for (int k = 0; k < KNN; ++k) {
            const int jj = idx3[k];
            const float dx = px - s_pos[3 * jj + 0];   // exact recompute from LDS
            const float dy = py - s_pos[3 * jj + 1];
            const float dz = pz - s_pos[3 * jj + 2];
            const float d2 = dx * dx + dy * dy + dz * dz;
            const float w  = 1.0f / fmaxf(d2, EPSV);
            s_i[row][k] = jj;
            s_w[row][k] = w;
            wsum += w;
        }
        s_w[row][3] = 1.0f / wsum;
    }
    __syncthreads();

    // ---- Feature gather + inverse-distance blend (float4, all 128 threads) ----
    const int nf4 = F >> 2;
    const int total = 16 * nf4;
    const float4* __restrict__ x4 = (const float4*)x;
    float4* __restrict__ out4 = (float4*)out;
    for (int idx = tid; idx < total; idx += NT) {
        const int m = idx / nf4;
        const int c = idx - m * nf4;
        const int   i0 = s_i[m][0], i1 = s_i[m][1], i2 = s_i[m][2];
        const float w0 = s_w[m][0], w1 = s_w[m][1], w2 = s_w[m][2];
        const float inv = s_w[m][3];
        const float4 a = x4[(long)i0 * nf4 + c];
        const float4 b = x4[(long)i1 * nf4 + c];
        const float4 d = x4[(long)i2 * nf4 + c];
        float4 o;
        o.x = (w0 * a.x + w1 * b.x + w2 * d.x) * inv;
        o.y = (w0 * a.y + w1 * b.y + w2 * d.y) * inv;
        o.z = (w0 * a.z + w1 * b.z + w2 * d.z) * inv;
        o.w = (w0 * a.w + w1 * b.w + w2 * d.w) * inv;
        out4[(long)(rowBase + m) * nf4 + c] = o;
    }
}

extern "C" void kernel_launch(void* const* d_in, const int* in_sizes, int n_in,
                              void* d_out, int out_size, void* d_ws, size_t ws_size,
                              hipStream_t stream) {
    const float* x     = (const float*)d_in[0];
    const float* pos_l = (const float*)d_in[1];
    const float* pos_h = (const float*)d_in[2];
    float* out = (float*)d_out;

    const int n_l = in_sizes[1] / 3;     // 4096
    const int n_h = in_sizes[2] / 3;     // 16384
    const int F   = in_sizes[0] / n_l;   // 256

    const int nblocks = n_h / 16;        // 16 fine rows per block (4 waves)
    knn_interp_wmma<<<dim3(nblocks), dim3(NT), 0, stream>>>(
        x, pos_l, pos_h, out, n_l, n_h, F);
}